// PixelWiseRNN_4612794875906
// MI455X (gfx1250) — compile-verified
//
#include <hip/hip_runtime.h>

// Per-pixel scalar tanh-RNN over T=256 steps. Pure bandwidth problem
// (512 MB stream, ~22us floor at 23.3 TB/s). Strategy: deep async
// global->LDS DMA ring (gfx1250 ASYNC path) for ~15 MB of in-flight
// loads, v_tanh_f32 for the nonlinearity, NT stores for the output.

typedef __attribute__((ext_vector_type(4))) float float4v;

#define BDIM 256
#define NBUF 16  // async ring depth; 16 * 256 threads * 16B = 64 KB LDS

__device__ __forceinline__ float fast_tanh(float v) {
#if __has_builtin(__builtin_amdgcn_tanhf)
    return __builtin_amdgcn_tanhf(v);   // gfx1250 v_tanh_f32 (TRANS op)
#else
    return tanhf(v);
#endif
}

__global__ __launch_bounds__(BDIM) void pixel_rnn_kernel(
    const float* __restrict__ x,
    const float* __restrict__ w_ih,
    const float* __restrict__ w_hh,
    const float* __restrict__ b_ih,
    const float* __restrict__ b_hh,
    float* __restrict__ out)
{
    constexpr int T = 256;
    constexpr int P = 16 * 64 * 64;        // Z*H*W = 65536 (pixels per (b,t) slab)

    __shared__ float4v smem[NBUF * BDIM];  // 64 KB ring, one 16B slot per thread per buffer

    const unsigned tid  = threadIdx.x;
    const unsigned gtid = blockIdx.x * BDIM + tid;
    const unsigned pix  = gtid * 4;        // 4 adjacent-w pixels per thread
    const unsigned b    = pix >> 16;       // pix / P
    const unsigned zhw  = pix & (P - 1);   // pix % P, 16B aligned (W=64 % 4 == 0)

    // Per-pixel parameters (read once, cached; tiny footprint)
    const float4v wih = *(const float4v*)(w_ih + zhw);
    const float4v whh = *(const float4v*)(w_hh + zhw);
    float4v bias      = *(const float4v*)(b_ih + zhw);
    {
        const float4v bhh = *(const float4v*)(b_hh + zhw);
        bias.x += bhh.x; bias.y += bhh.y; bias.z += bhh.z; bias.w += bhh.w;
    }

    const float* gx = x   + (size_t)b * T * P + zhw;   // stride P per t
    float*       go = out + (size_t)b * T * P + zhw;

    // ---- prologue: launch NBUF-1 async DMA loads (slots 0..NBUF-2) ----
#pragma unroll
    for (int i = 0; i < NBUF - 1; ++i) {
        unsigned loff = (unsigned)(size_t)&smem[i * BDIM + tid];          // LDS byte offset
        unsigned long long ga = (unsigned long long)(const void*)(gx + (size_t)i * P);
        asm volatile("global_load_async_to_lds_b128 %0, %1, off"
                     :: "v"(loff), "v"(ga) : "memory");
    }

    float4v h = {0.f, 0.f, 0.f, 0.f};      // h0 = 0

    // ---- main loop: t = 0 .. T-NBUF, prefetch distance NBUF-1 ----
    for (int t = 0; t <= T - NBUF; ++t) {
        // Ensure last iteration's LDS read has landed in VGPRs before
        // its ring slot (same slot we now overwrite) is re-targeted.
        asm volatile("s_wait_dscnt 0x0" ::: "memory");
        {
            const int tpre = t + NBUF - 1;
            unsigned loff = (unsigned)(size_t)&smem[(tpre & (NBUF - 1)) * BDIM + tid];
            unsigned long long ga = (unsigned long long)(const void*)(gx + (size_t)tpre * P);
            asm volatile("global_load_async_to_lds_b128 %0, %1, off"
                         :: "v"(loff), "v"(ga) : "memory");
        }
        // Async completions are in-order per wave: <= NBUF-1 outstanding
        // guarantees the load for step t has fully written LDS.
        asm volatile("s_wait_asynccnt 15" ::: "memory");

        const float4v xv = smem[(t & (NBUF - 1)) * BDIM + tid];

        h.x = fast_tanh(__builtin_fmaf(wih.x, xv.x, __builtin_fmaf(whh.x, h.x, bias.x)));
        h.y = fast_tanh(__builtin_fmaf(wih.y, xv.y, __builtin_fmaf(whh.y, h.y, bias.y)));
        h.z = fast_tanh(__builtin_fmaf(wih.z, xv.z, __builtin_fmaf(whh.z, h.z, bias.z)));
        h.w = fast_tanh(__builtin_fmaf(wih.w, xv.w, __builtin_fmaf(whh.w, h.w, bias.w)));

        __builtin_nontemporal_store(h, (float4v*)(go + (size_t)t * P));
    }

    // ---- tail: all remaining DMAs are already issued; drain once ----
    asm volatile("s_wait_asynccnt 0x0" ::: "memory");
#pragma unroll
    for (int t = T - NBUF + 1; t < T; ++t) {
        const float4v xv = smem[(t & (NBUF - 1)) * BDIM + tid];

        h.x = fast_tanh(__builtin_fmaf(wih.x, xv.x, __builtin_fmaf(whh.x, h.x, bias.x)));
        h.y = fast_tanh(__builtin_fmaf(wih.y, xv.y, __builtin_fmaf(whh.y, h.y, bias.y)));
        h.z = fast_tanh(__builtin_fmaf(wih.z, xv.z, __builtin_fmaf(whh.z, h.z, bias.z)));
        h.w = fast_tanh(__builtin_fmaf(wih.w, xv.w, __builtin_fmaf(whh.w, h.w, bias.w)));

        __builtin_nontemporal_store(h, (float4v*)(go + (size_t)t * P));
    }
}

extern "C" void kernel_launch(void* const* d_in, const int* in_sizes, int n_in,
                              void* d_out, int out_size, void* d_ws, size_t ws_size,
                              hipStream_t stream) {
    const float* x    = (const float*)d_in[0];
    const float* w_ih = (const float*)d_in[1];
    const float* w_hh = (const float*)d_in[2];
    const float* b_ih = (const float*)d_in[3];
    const float* b_hh = (const float*)d_in[4];
    float* out = (float*)d_out;

    // 262,144 pixels / 4 per thread = 65,536 threads = 256 blocks x 256
    dim3 grid(256), block(BDIM);
    hipLaunchKernelGGL(pixel_rnn_kernel, grid, block, 0, stream,
                       x, w_ih, w_hh, b_ih, b_hh, out);
}